// Custom_RNN_10101763080784
// MI455X (gfx1250) — compile-verified
//
#include <hip/hip_runtime.h>

#define NSTATE 200   // N = 2*GRID_SIZE

typedef __attribute__((ext_vector_type(2))) float v2f;
typedef __attribute__((ext_vector_type(8))) float v8f;

__device__ __forceinline__ float fsigmoid(float x) {
    return 1.0f / (1.0f + __expf(-x));
}
// overflow-safe tanh via hardware exp
__device__ __forceinline__ float ftanh(float x) {
    float e = __expf(2.0f * x);
    return 1.0f - 2.0f / (e + 1.0f);
}

__global__ __launch_bounds__(416) void rnn_fused(
    const float* __restrict__ state_space, const float* __restrict__ inputs,
    const float* __restrict__ w_in,  const float* __restrict__ b_in,
    const float* __restrict__ yw_i,  const float* __restrict__ yw_h,
    const float* __restrict__ yb,
    const float* __restrict__ bw_i,  const float* __restrict__ bw_h,
    const float* __restrict__ bb,
    const float* __restrict__ w_us,  const float* __restrict__ b_us,
    const float* __restrict__ w_v,   const float* __restrict__ b_v,
    float* __restrict__ out)
{
    __shared__ float s_us[NSTATE];        // us_flat
    __shared__ float s_xcat[2 * NSTATE];  // concat(us_flat, o1)
    __shared__ float s_t[1];

    const int tid = threadIdx.x;

    // ---------------- Phase 1: sequential scans (wave 0, lanes replicated) ----
    if (tid < 32) {
        float in0 = inputs[0], in1 = inputs[1], in2 = inputs[2];
        float tval = (in0 == 1.0f) ? 0.0f
                   : (in1 == 1.0f) ? 1.0f
                   : (in2 == 1.0f) ? 2.0f : 0.0f;
        if (tid == 0) s_t[0] = tval;

        // h0 = inputs @ w_in + b_in     (w_in row-major (3,2))
        float h0 = in0 * w_in[0] + in1 * w_in[2] + in2 * w_in[4] + b_in[0];
        float h1 = in0 * w_in[1] + in1 * w_in[3] + in2 * w_in[5] + b_in[1];

        // ---- yellow scan (H=2); att = running column-sum of ys (softmax over
        //      size-1 axis == 1.0, so attention collapses to a sum) ----
        float att0 = 0.0f, att1 = 0.0f;
        for (int n = 0; n < NSTATE; ++n) {
            const float* wi = yw_i + n * 12;  // (2,6)
            const float* wh = yw_h + n * 12;  // (2,6)
            const float* b  = yb   + n * 6;
            if (n + 1 < NSTATE) {             // hide next step's miss latency
                __builtin_prefetch(yw_i + (n + 1) * 12, 0, 0);
                __builtin_prefetch(yw_h + (n + 1) * 12, 0, 0);
            }
            float x0 = state_space[2 * n], x1 = state_space[2 * n + 1];
            float gx[6];
            #pragma unroll
            for (int j = 0; j < 6; ++j) gx[j] = x0 * wi[j] + x1 * wi[6 + j];
            float z0 = fsigmoid(gx[0] + h0 * wh[0] + h1 * wh[6] + b[0]);
            float z1 = fsigmoid(gx[1] + h0 * wh[1] + h1 * wh[7] + b[1]);
            float r0 = fsigmoid(gx[2] + h0 * wh[2] + h1 * wh[8] + b[2]);
            float r1 = fsigmoid(gx[3] + h0 * wh[3] + h1 * wh[9] + b[3]);
            float rh0 = r0 * h0, rh1 = r1 * h1;   // (r*h) @ wh[:,4:]
            float a0 = ftanh(gx[4] + rh0 * wh[4] + rh1 * wh[10] + b[4]);
            float a1 = ftanh(gx[5] + rh0 * wh[5] + rh1 * wh[11] + b[5]);
            h0 = (1.0f - z0) * h0 + z0 * a0;
            h1 = (1.0f - z1) * h1 + z1 * a1;
            att0 += h0; att1 += h1;
        }

        // ---- blue scan (hid = 2H = 4) ----
        float d[4] = {h0, h1, 0.0f, 0.0f};   // c0 = concat(hn, zeros)
        float u0 = 0.0f, u1v = 0.0f;
        float wus0 = w_us[0], wus1 = w_us[1], bus = b_us[0];
        for (int n = 0; n < NSTATE; ++n) {
            const float* wi = bw_i + n * 48;  // (4,12)
            const float* wh = bw_h + n * 48;  // (4,12)
            const float* b  = bb   + n * 12;
            if (n + 1 < NSTATE) {
                __builtin_prefetch(bw_i + (n + 1) * 48, 0, 0);
                __builtin_prefetch(bw_h + (n + 1) * 48, 0, 0);
            }
            float s0 = att0, s1 = att1, s2 = u0, s3 = u1v;  // h of GRU
            float gx[12];
            #pragma unroll
            for (int j = 0; j < 12; ++j)     // d0 @ bw_i
                gx[j] = d[0]*wi[j] + d[1]*wi[12+j] + d[2]*wi[24+j] + d[3]*wi[36+j];
            float z[4], r[4];
            #pragma unroll
            for (int j = 0; j < 4; ++j)
                z[j] = fsigmoid(gx[j]   + s0*wh[j]    + s1*wh[12+j] +
                                s2*wh[24+j] + s3*wh[36+j] + b[j]);
            #pragma unroll
            for (int j = 0; j < 4; ++j)
                r[j] = fsigmoid(gx[4+j] + s0*wh[4+j]  + s1*wh[16+j] +
                                s2*wh[28+j] + s3*wh[40+j] + b[4+j]);
            float rs0 = r[0]*s0, rs1 = r[1]*s1, rs2 = r[2]*s2, rs3 = r[3]*s3;
            float sv[4] = {s0, s1, s2, s3};
            float nh[4];
            #pragma unroll
            for (int j = 0; j < 4; ++j) {
                float a = ftanh(gx[8+j] + rs0*wh[8+j] + rs1*wh[20+j] +
                                rs2*wh[32+j] + rs3*wh[44+j] + b[8+j]);
                nh[j] = (1.0f - z[j]) * sv[j] + z[j] * a;
            }
            d[0]=nh[0]; d[1]=nh[1]; d[2]=nh[2]; d[3]=nh[3];
            u0 = nh[0]; u1v = nh[1];
            if (tid == 0) s_us[n] = nh[0]*wus0 + nh[1]*wus1 + bus;  // us@w_us+b
        }
    }
    __syncthreads();

    // ---------------- Phase 2: pointwise c / o1 / concat vector --------------
    if (tid < NSTATE) {
        float tval = s_t[0];
        float X = state_space[2 * tid], o = state_space[2 * tid + 1];
        float usf = s_us[tid];
        float cu = X * fsigmoid(usf * 0.001f);
        float c  = (tval >= 2.0f) ? X : cu;
        float o1 = fminf(((X - c >= 0.2f) ? 1.0f : 0.0f) + o, 1.0f);
        s_xcat[tid]          = usf;
        s_xcat[NSTATE + tid] = o1;
        out[2 * tid + 1] = c;                 // second output column
    }
    __syncthreads();

    // ---------------- Phase 3: vu = xcat(1x400) @ w_v(400x200) + b_v ---------
    // V_WMMA_F32_16X16X4_F32, input vector in row M=0 of A. 13 waves, one
    // 16-wide output tile each, 100 K-steps of 4, B loads double-buffered.
    //
    // Out-of-range columns (tile 12, col>=200) only pollute D columns we never
    // store, so the address clamp alone is sufficient -- no value masking.
    const int wave = tid >> 5, lane = tid & 31;
    const int l = lane & 15;
    const bool hi = lane >= 16;
    const int n0 = wave * 16, col = n0 + l;
    const int colc = (col < NSTATE) ? col : (NSTATE - 1);
    const int koff = hi ? 2 : 0;              // A/B K-row pair for this half
    const float amask = (l == 0) ? 1.0f : 0.0f;  // only M=0 row of A is real
    const float* wp = w_v + colc;             // column pointer into w_v

    v8f acc = {};
    acc[0] = (!hi) ? b_v[colc] : 0.0f;        // C row M=0 -> bias

    // prime the B double-buffer (K rows koff, koff+1 of step 0)
    v2f B = { wp[koff * NSTATE], wp[(koff + 1) * NSTATE] };

    for (int k0 = 0; k0 < 2 * NSTATE; k0 += 4) {
        const int kr = k0 + koff;
        // issue next iteration's B loads first (clamped so always in-range;
        // the extra values of the final iteration are never consumed)
        const int krn = ((k0 + 4 < 2 * NSTATE) ? (k0 + 4) : k0) + koff;
        v2f Bn = { wp[krn * NSTATE], wp[(krn + 1) * NSTATE] };
        // A: uniform LDS broadcast per lane-half, masked to row M=0 (no EXEC
        // divergence -- plain ds_load + v_mul)
        v2f A = { s_xcat[kr] * amask, s_xcat[kr + 1] * amask };
        acc = __builtin_amdgcn_wmma_f32_16x16x4_f32(
            /*neg_a=*/false, A, /*neg_b=*/false, B,
            /*c_mod=*/(short)0, acc, /*reuse_a=*/false, /*reuse_b=*/false);
        B = Bn;
    }

    if (!hi && col < NSTATE) {
        float tval = s_t[0];
        float vu = acc[0];                    // D[0][col]
        out[2 * col] = (tval > 2.0f) ? 0.0f : vu;  // first output column
    }
}

extern "C" void kernel_launch(void* const* d_in, const int* in_sizes, int n_in,
                              void* d_out, int out_size, void* d_ws, size_t ws_size,
                              hipStream_t stream) {
    (void)in_sizes; (void)n_in; (void)out_size; (void)d_ws; (void)ws_size;
    const float* state_space = (const float*)d_in[0];
    const float* inputs      = (const float*)d_in[1];
    const float* w_in        = (const float*)d_in[2];
    const float* b_in        = (const float*)d_in[3];
    const float* yw_i        = (const float*)d_in[4];
    const float* yw_h        = (const float*)d_in[5];
    const float* yb          = (const float*)d_in[6];
    // d_in[7] = aw, d_in[8] = ab : dead (softmax over size-1 axis == 1)
    const float* bw_i        = (const float*)d_in[9];
    const float* bw_h        = (const float*)d_in[10];
    const float* bb          = (const float*)d_in[11];
    const float* w_us        = (const float*)d_in[12];
    const float* b_us        = (const float*)d_in[13];
    const float* w_v         = (const float*)d_in[14];
    const float* b_v         = (const float*)d_in[15];
    float* out = (float*)d_out;

    rnn_fused<<<dim3(1), dim3(416), 0, stream>>>(
        state_space, inputs, w_in, b_in, yw_i, yw_h, yb,
        bw_i, bw_h, bb, w_us, b_us, w_v, b_v, out);
}